// GCN_16887811408655
// MI455X (gfx1250) — compile-verified
//
#include <hip/hip_runtime.h>

typedef __attribute__((ext_vector_type(2))) float v2f;
typedef __attribute__((ext_vector_type(8))) float v8f;

static constexpr int NN  = 20000;   // nodes
static constexpr int FIN = 512;     // input features
static constexpr int HD  = 256;     // hidden
static constexpr int CD  = 40;      // classes
static constexpr int EE  = 640000;  // edges

// ---------------- degree / normalization ----------------

__global__ __launch_bounds__(256) void k_zero(float* __restrict__ p, int n) {
  int i = blockIdx.x * 256 + threadIdx.x;
  if (i < n) p[i] = 0.0f;
}

__global__ __launch_bounds__(256) void k_degree(const int* __restrict__ dst,
                                                float* __restrict__ deg, int e) {
  int i = blockIdx.x * 256 + threadIdx.x;
  if (i < e) unsafeAtomicAdd(&deg[dst[i]], 1.0f);
}

__global__ __launch_bounds__(256) void k_dinv(float* __restrict__ d, int n) {
  int i = blockIdx.x * 256 + threadIdx.x;
  if (i < n) d[i] = rsqrtf(d[i] + 1.0f);
}

// ---------------- f32 WMMA GEMM: Y[NN,HD] = X[NN,FIN] @ W[FIN,HD] ----------------
// Each wave: 16 rows x 64 cols (4 accumulators), K stepped by 4 via
// V_WMMA_F32_16X16X4_F32. Block = 256 threads = 8 waves = 32 rows x 256 cols.
__global__ __launch_bounds__(256) void k_gemm1(const float* __restrict__ X,
                                               const float* __restrict__ W,
                                               float* __restrict__ Y) {
  const int lane  = threadIdx.x & 31;
  const int wave  = threadIdx.x >> 5;
  const int mtile = blockIdx.x * 2 + (wave >> 2);
  const int nbase = (wave & 3) * 64;
  const int half  = lane >> 4;
  const int l15   = lane & 15;
  const int row   = mtile * 16 + l15;
  const int ncol  = nbase + l15;

  v8f a0 = {}, a1 = {}, a2 = {}, a3 = {};
  const float* xr = X + (size_t)row * FIN;

  for (int kk = 0; kk < FIN; kk += 4) {
    const int k0 = kk + 2 * half;           // A: VGPR0->K=k0, VGPR1->K=k0+1
    v2f a; a.x = xr[k0]; a.y = xr[k0 + 1];
    const float* w0 = W + (size_t)k0 * HD + ncol;
    const float* w1 = w0 + HD;
    v2f b0; b0.x = w0[0];  b0.y = w1[0];
    v2f b1; b1.x = w0[16]; b1.y = w1[16];
    v2f b2; b2.x = w0[32]; b2.y = w1[32];
    v2f b3; b3.x = w0[48]; b3.y = w1[48];
    a0 = __builtin_amdgcn_wmma_f32_16x16x4_f32(false, a, false, b0, (short)0, a0, false, false);
    a1 = __builtin_amdgcn_wmma_f32_16x16x4_f32(false, a, false, b1, (short)0, a1, false, false);
    a2 = __builtin_amdgcn_wmma_f32_16x16x4_f32(false, a, false, b2, (short)0, a2, false, false);
    a3 = __builtin_amdgcn_wmma_f32_16x16x4_f32(false, a, false, b3, (short)0, a3, false, false);
  }

  const int mbase = mtile * 16 + half * 8;  // C/D: lanes16-31 hold M+8
#pragma unroll
  for (int r = 0; r < 8; ++r) {
    float* yr = Y + (size_t)(mbase + r) * HD + nbase + l15;
    yr[0]  = a0[r];
    yr[16] = a1[r];
    yr[32] = a2[r];
    yr[48] = a3[r];
  }
}

// Y[NN,CD] = X[NN,HD] @ W[HD,CD], CD=40 padded to 48 (3 col tiles, guarded)
__global__ __launch_bounds__(256) void k_gemm2(const float* __restrict__ X,
                                               const float* __restrict__ W,
                                               float* __restrict__ Y) {
  const int lane  = threadIdx.x & 31;
  const int wave  = threadIdx.x >> 5;
  const int mtile = blockIdx.x * 8 + wave;
  if (mtile >= NN / 16) return;             // wave-uniform: EXEC stays all-1s
  const int half = lane >> 4;
  const int l15  = lane & 15;
  const int row  = mtile * 16 + l15;

  v8f a0 = {}, a1 = {}, a2 = {};
  const float* xr = X + (size_t)row * HD;

  for (int kk = 0; kk < HD; kk += 4) {
    const int k0 = kk + 2 * half;
    v2f a; a.x = xr[k0]; a.y = xr[k0 + 1];
    const float* w0 = W + (size_t)k0 * CD;
    const float* w1 = w0 + CD;
    v2f b0; b0.x = w0[l15];      b0.y = w1[l15];
    v2f b1; b1.x = w0[16 + l15]; b1.y = w1[16 + l15];
    const int c2 = 32 + l15;
    v2f b2; b2.x = (c2 < CD) ? w0[c2] : 0.0f;
            b2.y = (c2 < CD) ? w1[c2] : 0.0f;
    a0 = __builtin_amdgcn_wmma_f32_16x16x4_f32(false, a, false, b0, (short)0, a0, false, false);
    a1 = __builtin_amdgcn_wmma_f32_16x16x4_f32(false, a, false, b1, (short)0, a1, false, false);
    a2 = __builtin_amdgcn_wmma_f32_16x16x4_f32(false, a, false, b2, (short)0, a2, false, false);
  }

  const int mbase = mtile * 16 + half * 8;
#pragma unroll
  for (int r = 0; r < 8; ++r) {
    float* yr = Y + (size_t)(mbase + r) * CD;
    yr[l15]      = a0[r];
    yr[16 + l15] = a1[r];                   // 16..31 < 40 always
    if (32 + l15 < CD) yr[32 + l15] = a2[r];
  }
}

// ---------------- aggregation (gather / atomic scatter) ----------------

// agg[node] = hw[node] * dinv[node]^2   (self-loop term as initializer)
template <int F, int F4>
__global__ __launch_bounds__(256) void k_init_agg(const float* __restrict__ hw,
                                                  const float* __restrict__ dinv,
                                                  float* __restrict__ agg, int n) {
  int i = blockIdx.x * 256 + threadIdx.x;
  if (i >= n * F4) return;
  int node = i / F4;
  int j = (i - node * F4) * 4;
  float di = dinv[node];
  float s = di * di;
  float4 v = *(const float4*)(hw + (size_t)node * F + j);
  float4 o = make_float4(v.x * s, v.y * s, v.z * s, v.w * s);
  *(float4*)(agg + (size_t)node * F + j) = o;
}

// agg[dst] += hw[src] * dinv[src]*dinv[dst]   (hardware f32 atomics)
template <int F, int F4>
__global__ __launch_bounds__(256) void k_scatter(const float* __restrict__ hw,
                                                 const float* __restrict__ dinv,
                                                 const int* __restrict__ src,
                                                 const int* __restrict__ dst,
                                                 float* __restrict__ agg, int e) {
  int i = blockIdx.x * 256 + threadIdx.x;
  if (i >= e * F4) return;
  int eid = i / F4;
  int j = (i - eid * F4) * 4;
  int s = src[eid], d = dst[eid];
  float coef = dinv[s] * dinv[d];
  float4 v = *(const float4*)(hw + (size_t)s * F + j);
  float* ap = agg + (size_t)d * F + j;
  unsafeAtomicAdd(ap + 0, v.x * coef);
  unsafeAtomicAdd(ap + 1, v.y * coef);
  unsafeAtomicAdd(ap + 2, v.z * coef);
  unsafeAtomicAdd(ap + 3, v.w * coef);
}

template <int F, bool RELU>
__global__ __launch_bounds__(256) void k_bias_act(float* __restrict__ a,
                                                  const float* __restrict__ b, int n) {
  int i = blockIdx.x * 256 + threadIdx.x;
  if (i >= n * F) return;
  int f = i % F;
  float v = a[i] + b[f];
  if (RELU) v = fmaxf(v, 0.0f);
  a[i] = v;
}

// ---------------- launch ----------------

extern "C" void kernel_launch(void* const* d_in, const int* in_sizes, int n_in,
                              void* d_out, int out_size, void* d_ws, size_t ws_size,
                              hipStream_t stream) {
  (void)in_sizes; (void)n_in; (void)out_size; (void)ws_size;
  const float* x  = (const float*)d_in[0];
  const int*   ei = (const int*)  d_in[1];
  const float* W1 = (const float*)d_in[2];
  const float* b1 = (const float*)d_in[3];
  const float* W2 = (const float*)d_in[4];
  const float* b2 = (const float*)d_in[5];
  float* out = (float*)d_out;

  float* ws   = (float*)d_ws;
  float* dinv = ws;                              // NN
  float* hw1  = dinv + NN;                       // NN*HD
  float* h1   = hw1 + (size_t)NN * HD;           // NN*HD (agg1 -> relu'd hidden)
  float* hw2  = h1  + (size_t)NN * HD;           // NN*CD

  const int* src = ei;
  const int* dst = ei + EE;

  const int T = 256;
  auto blk = [](int n) { return (n + 255) / 256; };

  // normalization: deg -> dinv
  k_zero<<<blk(NN), T, 0, stream>>>(dinv, NN);
  k_degree<<<blk(EE), T, 0, stream>>>(dst, dinv, EE);
  k_dinv<<<blk(NN), T, 0, stream>>>(dinv, NN);

  // layer 1: hw1 = x @ W1  (WMMA f32), then normalized aggregation + bias + relu
  k_gemm1<<<NN / 32, T, 0, stream>>>(x, W1, hw1);
  k_init_agg<HD, HD / 4><<<blk(NN * (HD / 4)), T, 0, stream>>>(hw1, dinv, h1, NN);
  k_scatter<HD, HD / 4><<<blk(EE / 16 * (HD / 4) * 16 / 16 * 16) /* EE*64 */, T, 0, stream>>>(
      hw1, dinv, src, dst, h1, EE);
  k_bias_act<HD, true><<<blk(NN * HD), T, 0, stream>>>(h1, b1, NN);

  // layer 2: hw2 = h1 @ W2 (WMMA f32), aggregation directly into d_out
  k_gemm2<<<(NN / 16 + 7) / 8, T, 0, stream>>>(h1, W2, hw2);
  k_init_agg<CD, CD / 4><<<blk(NN * (CD / 4)), T, 0, stream>>>(hw2, dinv, out, NN);
  k_scatter<CD, CD / 4><<<blk(EE * (CD / 4)), T, 0, stream>>>(hw2, dinv, src, dst, out, EE);
  k_bias_act<CD, false><<<blk(NN * CD), T, 0, stream>>>(out, b2, NN);
  // TEMPERATURE == 1.0 -> division is identity
}